// IterBlock_86887188398978
// MI455X (gfx1250) — compile-verified
//
#include <hip/hip_runtime.h>

typedef float v4f __attribute__((ext_vector_type(4)));
typedef unsigned int u32x4 __attribute__((ext_vector_type(4)));
typedef int i32x4 __attribute__((ext_vector_type(4)));
typedef int i32x8 __attribute__((ext_vector_type(8)));

namespace {
constexpr float L_Tc   = 0.045f;
constexpr float TAUTc  = 2.5f;
constexpr float THETAc = 0.3f;
constexpr float EPSc   = 1e-12f;

constexpr int Bn = 16, Hn = 512, Wn = 512;
constexpr int TSX = 64, TSY = 16;
constexpr int CX  = TSX / 4;   // 16 interior float4 chunks per tile row
constexpr int EXC = CX + 1;    // 17 chunks incl. +x halo
constexpr int EXR = TSY + 1;   // 17 rows incl. +y halo
constexpr int PR  = TSY + 2;   // 18 rows of staged p12/p22 (y-1 .. y+TSY)
constexpr int LDW = TSX + 4;   // 68 floats/row (272B stride; == TDM tile_dim0)
}

// One 2D TDM descriptor load: tile_dim0 = LDW(=68) elements x `tile1` rows,
// row stride Wn, 4-byte elements, written contiguously into LDS (row stride
// == tile_dim0 == LDW, matching the static LDS arrays). x/y beyond
// tensor_dim0/1 (distance from the tile origin to the image edge) zero-fill.
__device__ __forceinline__ void tdm_load_2d(const void* gsrc, unsigned ldsoff,
                                            int tdim0, int tdim1, int tile1) {
#if __has_builtin(__builtin_amdgcn_tensor_load_to_lds)
  unsigned long long ga = (unsigned long long)gsrc;
  u32x4 g0;
  g0[0] = 1u;                                   // count=1, user descriptor
  g0[1] = ldsoff;                               // lds_addr (bytes)
  g0[2] = (unsigned)(ga & 0xffffffffull);       // global_addr[31:0]
  g0[3] = (unsigned)((ga >> 32) & 0x1ffffffull) // global_addr[56:32]
          | (2u << 30);                         // type=2 ("image")
  i32x8 g1;
  g1[0] = (int)(2u << 16);                      // data_size=2 -> 4 bytes
  g1[1] = (int)(((unsigned)tdim0 & 0xffffu) << 16);           // tensor_dim0 lo
  g1[2] = (int)(((unsigned)tdim0 >> 16) |
                (((unsigned)tdim1 & 0xffffu) << 16));         // dim0 hi|dim1 lo
  g1[3] = (int)(((unsigned)tdim1 >> 16) |
                ((unsigned)LDW << 16));                       // dim1 hi|tile0
  g1[4] = tile1 & 0xffff;                                     // tile_dim1
  g1[5] = Wn;                                                 // dim0_stride lo
  g1[6] = 0;                                                  // stride hi
  g1[7] = 0;
  i32x4 gz = {0, 0, 0, 0};                      // groups 2/3: unused (2D)
# if __clang_major__ >= 23
  i32x8 gz8 = {0, 0, 0, 0, 0, 0, 0, 0};
  __builtin_amdgcn_tensor_load_to_lds(g0, g1, gz, gz, gz8, 0);
# else
  __builtin_amdgcn_tensor_load_to_lds(g0, g1, gz, gz, 0);
# endif
#else
  (void)gsrc; (void)ldsoff; (void)tdim0; (void)tdim1; (void)tile1;
#endif
}

__global__ __launch_bounds__(256) void tvl1_fused(
    const float* __restrict__ dxw, const float* __restrict__ dyw,
    const float* __restrict__ u1,  const float* __restrict__ u2,
    const float* __restrict__ grad,
    const float* __restrict__ p11, const float* __restrict__ p12,
    const float* __restrict__ p21, const float* __restrict__ p22,
    float* __restrict__ o_p11, float* __restrict__ o_p12,
    float* __restrict__ o_p21, float* __restrict__ o_p22,
    float* __restrict__ o_rho, float* __restrict__ o_u1, float* __restrict__ o_u2)
{
  // p12/p22: rows r <-> gy = ty0 + r - 1, cols j <-> gx = tx0 + j
  __shared__ __align__(16) float s_p12[PR][LDW];
  __shared__ __align__(16) float s_p22[PR][LDW];
  // p11/p21: rows r <-> gy = ty0 + r,     cols j <-> gx = tx0 + j
  __shared__ __align__(16) float s_p11[EXR][LDW];
  __shared__ __align__(16) float s_p21[EXR][LDW];
  // u1n/u2n: rows r <-> gy = ty0 + r,     cols j <-> gx = tx0 + j
  __shared__ __align__(16) float s_u1[EXR][LDW];
  __shared__ __align__(16) float s_u2[EXR][LDW];

  const int b   = blockIdx.z;
  const int tx0 = blockIdx.x * TSX;
  const int ty0 = blockIdx.y * TSY;
  const long base = (long)b * Hn * Wn;
  const int tid = threadIdx.y * 16 + threadIdx.x;

  // ---- Phase 0: stage all four p fields into LDS.
#if __has_builtin(__builtin_amdgcn_tensor_load_to_lds)
  // Tensor Data Mover path: 4 descriptors per tile, tracked by TENSORcnt.
  if (tid < 32) {  // wave 0 issues; TDM is per-wave, side-effecting (no if-cvt)
    const int rows1 = (EXR < Hn - ty0) ? EXR : (Hn - ty0);
    tdm_load_2d(p11 + base + (long)ty0 * Wn + tx0,
                (unsigned)(unsigned long long)&s_p11[0][0],
                Wn - tx0, Hn - ty0, rows1);
    tdm_load_2d(p21 + base + (long)ty0 * Wn + tx0,
                (unsigned)(unsigned long long)&s_p21[0][0],
                Wn - tx0, Hn - ty0, rows1);
    const int gy0 = (ty0 == 0) ? 0 : (ty0 - 1);
    const int r0  = (ty0 == 0) ? 1 : 0;
    const int rows2 = ((PR - r0) < Hn - gy0) ? (PR - r0) : (Hn - gy0);
    tdm_load_2d(p12 + base + (long)gy0 * Wn + tx0,
                (unsigned)(unsigned long long)&s_p12[r0][0],
                Wn - tx0, Hn - gy0, rows2);
    tdm_load_2d(p22 + base + (long)gy0 * Wn + tx0,
                (unsigned)(unsigned long long)&s_p22[r0][0],
                Wn - tx0, Hn - gy0, rows2);
  }
# if __has_builtin(__builtin_amdgcn_s_wait_tensorcnt)
  __builtin_amdgcn_s_wait_tensorcnt(0);
# else
  asm volatile("s_wait_tensorcnt 0" ::: "memory");
# endif
#else
  // Fallback: per-lane async global->LDS DMA (ASYNCcnt).
  for (int c = tid; c < 2 * PR * EXC; c += 256) {  // p12 / p22
    const int half = (c >= PR * EXC) ? 1 : 0;
    const int idx  = c - half * (PR * EXC);
    const int r  = idx / EXC;
    const int cc = idx - r * EXC;
    const int gy = ty0 + r - 1;
    const int gx = tx0 + cc * 4;
    if (gy >= 0 && gy < Hn && gx < Wn) {
      const float* src = (half ? p22 : p12) + base + (long)gy * Wn + gx;
      float (*sp)[LDW] = half ? s_p22 : s_p12;
      unsigned ldsoff = (unsigned)(unsigned long long)&sp[r][cc * 4];
      unsigned long long ga = (unsigned long long)src;
      asm volatile("global_load_async_to_lds_b128 %0, %1, off"
                   :: "v"(ldsoff), "v"(ga) : "memory");
    }
  }
  for (int c = tid; c < 2 * EXR * EXC; c += 256) {  // p11 / p21
    const int half = (c >= EXR * EXC) ? 1 : 0;
    const int idx  = c - half * (EXR * EXC);
    const int r  = idx / EXC;
    const int cc = idx - r * EXC;
    const int gy = ty0 + r;
    const int gx = tx0 + cc * 4;
    if (gy < Hn && gx < Wn) {
      const float* src = (half ? p21 : p11) + base + (long)gy * Wn + gx;
      float (*sp)[LDW] = half ? s_p21 : s_p11;
      unsigned ldsoff = (unsigned)(unsigned long long)&sp[r][cc * 4];
      unsigned long long ga = (unsigned long long)src;
      asm volatile("global_load_async_to_lds_b128 %0, %1, off"
                   :: "v"(ldsoff), "v"(ga) : "memory");
    }
  }
# if __has_builtin(__builtin_amdgcn_s_wait_asynccnt)
  __builtin_amdgcn_s_wait_asynccnt(0);
# else
  asm volatile("s_wait_asynccnt 0" ::: "memory");
# endif
#endif
  __syncthreads();

  // ---- Stage 1: rho / u1n / u2n over the extended (TSY+1) x (TSX+4) tile.
  for (int c = tid; c < EXR * EXC; c += 256) {
    const int ecy = c / EXC;
    const int ecx = c - ecy * EXC;
    const int gy = ty0 + ecy;
    const int gx = tx0 + ecx * 4;
    if (gy < Hn && gx < Wn) {
      const long off = base + (long)gy * Wn + gx;
      v4f vdx = *(const v4f*)(dxw + off);
      v4f vdy = *(const v4f*)(dyw + off);
      v4f vu1 = *(const v4f*)(u1 + off);
      v4f vu2 = *(const v4f*)(u2 + off);
      v4f vg  = *(const v4f*)(grad + off);
      v4f v11 = *(const v4f*)&s_p11[ecy][ecx * 4];
      v4f v21 = *(const v4f*)&s_p21[ecy][ecx * 4];
      float p11m, p21m;                 // value at gx-1 (x-halo)
      if (ecx > 0)      { p11m = s_p11[ecy][ecx * 4 - 1];
                          p21m = s_p21[ecy][ecx * 4 - 1]; }
      else if (gx > 0)  { p11m = p11[off - 1]; p21m = p21[off - 1]; }
      else              { p11m = 0.0f; p21m = 0.0f; }  // masked by xi==0
      v4f c12 = *(const v4f*)&s_p12[ecy + 1][ecx * 4];
      v4f m12 = *(const v4f*)&s_p12[ecy    ][ecx * 4];
      v4f c22 = *(const v4f*)&s_p22[ecy + 1][ecx * 4];
      v4f m22 = *(const v4f*)&s_p22[ecy    ][ecx * 4];
      v4f rr, r1, r2;
      #pragma unroll
      for (int i = 0; i < 4; ++i) {
        const int xi = gx + i;
        const float rho = vdx[i] * vu1[i] + vdy[i] * vu2[i] + EPSc;
        const float lg = L_Tc * vg[i];
        float coef;
        if (rho < -lg)         coef = L_Tc;          // m1
        else if (rho > lg)     coef = -L_Tc;         // m2
        else if (vg[i] > EPSc) coef = -rho / vg[i];  // m3 (g_safe == grad)
        else                   coef = 0.0f;
        const float v1v = coef * vdx[i] + vu1[i];
        const float v2v = coef * vdy[i] + vu2[i];
        const float l11 = (i > 0) ? v11[i - 1] : p11m;
        const float l21 = (i > 0) ? v21[i - 1] : p21m;
        const float dx1 = (xi == 0) ? v11[i] : ((xi == Wn - 1) ? -l11 : v11[i] - l11);
        const float dx2 = (xi == 0) ? v21[i] : ((xi == Wn - 1) ? -l21 : v21[i] - l21);
        const float dy1 = (gy == 0) ? c12[i] : ((gy == Hn - 1) ? -m12[i] : c12[i] - m12[i]);
        const float dy2 = (gy == 0) ? c22[i] : ((gy == Hn - 1) ? -m22[i] : c22[i] - m22[i]);
        rr[i] = rho;
        r1[i] = v1v + THETAc * (dx1 + dy1);
        r2[i] = v2v + THETAc * (dx2 + dy2);
      }
      *(v4f*)&s_u1[ecy][ecx * 4] = r1;
      *(v4f*)&s_u2[ecy][ecx * 4] = r2;
      if (ecy < TSY && ecx < CX) {  // interior: stream out, never re-read
        __builtin_nontemporal_store(rr, (v4f*)(o_rho + off));
        __builtin_nontemporal_store(r1, (v4f*)(o_u1 + off));
        __builtin_nontemporal_store(r2, (v4f*)(o_u2 + off));
      }
    }
  }
  __syncthreads();

  // ---- Stage 2: forward gradients of u1n/u2n + dual update (pure LDS in,
  // non-temporal b128 out).
  {
    const int cx = threadIdx.x * 4;
    const int ty = threadIdx.y;
    const int gy = ty0 + ty;
    const int gx = tx0 + cx;
    const long off = base + (long)gy * Wn + gx;
    v4f a1 = *(const v4f*)&s_u1[ty][cx];
    v4f b1 = *(const v4f*)&s_u1[ty + 1][cx];
    v4f a2 = *(const v4f*)&s_u2[ty][cx];
    v4f b2 = *(const v4f*)&s_u2[ty + 1][cx];
    const float a1n = s_u1[ty][cx + 4];
    const float a2n = s_u2[ty][cx + 4];
    v4f q11 = *(const v4f*)&s_p11[ty][cx];
    v4f q21 = *(const v4f*)&s_p21[ty][cx];
    v4f q12 = *(const v4f*)&s_p12[ty + 1][cx];
    v4f q22 = *(const v4f*)&s_p22[ty + 1][cx];
    v4f o11, o12, o21, o22;
    #pragma unroll
    for (int i = 0; i < 4; ++i) {
      const int xi = gx + i;
      const float u1r = (i < 3) ? a1[i + 1] : a1n;
      const float u2r = (i < 3) ? a2[i + 1] : a2n;
      const float u1x = (xi < Wn - 1) ? u1r - a1[i] : 0.0f;
      const float u1y = (gy < Hn - 1) ? b1[i] - a1[i] : 0.0f;
      const float u2x = (xi < Wn - 1) ? u2r - a2[i] : 0.0f;
      const float u2y = (gy < Hn - 1) ? b2[i] - a2[i] : 0.0f;
      const float n1 = 1.0f + TAUTc * sqrtf(u1x * u1x + u1y * u1y + EPSc);
      const float n2 = 1.0f + TAUTc * sqrtf(u2x * u2x + u2y * u2y + EPSc);
      o11[i] = (q11[i] + TAUTc * u1x) / n1;
      o12[i] = (q12[i] + TAUTc * u1y) / n1;
      o21[i] = (q21[i] + TAUTc * u2x) / n2;
      o22[i] = (q22[i] + TAUTc * u2y) / n2;
    }
    __builtin_nontemporal_store(o11, (v4f*)(o_p11 + off));
    __builtin_nontemporal_store(o12, (v4f*)(o_p12 + off));
    __builtin_nontemporal_store(o21, (v4f*)(o_p21 + off));
    __builtin_nontemporal_store(o22, (v4f*)(o_p22 + off));
  }
}

extern "C" void kernel_launch(void* const* d_in, const int* in_sizes, int n_in,
                              void* d_out, int out_size, void* d_ws, size_t ws_size,
                              hipStream_t stream) {
  const float* dxw  = (const float*)d_in[0];  // diff2_x_warp
  const float* dyw  = (const float*)d_in[1];  // diff2_y_warp
  const float* u1   = (const float*)d_in[2];
  const float* u2   = (const float*)d_in[3];
  const float* grad = (const float*)d_in[4];
  const float* p11  = (const float*)d_in[5];
  const float* p12  = (const float*)d_in[6];
  const float* p21  = (const float*)d_in[7];
  const float* p22  = (const float*)d_in[8];

  float* out = (float*)d_out;
  const long N = (long)Bn * Hn * Wn;
  float* o_p11 = out + 0 * N;
  float* o_p12 = out + 1 * N;
  float* o_p21 = out + 2 * N;
  float* o_p22 = out + 3 * N;
  float* o_rho = out + 4 * N;
  float* o_u1  = out + 5 * N;
  float* o_u2  = out + 6 * N;

  dim3 grid(Wn / TSX, Hn / TSY, Bn);   // 8 x 32 x 16 = 4096 blocks
  dim3 block(16, 16, 1);               // 256 threads = 8 wave32
  tvl1_fused<<<grid, block, 0, stream>>>(
      dxw, dyw, u1, u2, grad, p11, p12, p21, p22,
      o_p11, o_p12, o_p21, o_p22, o_rho, o_u1, o_u2);
}